// WaveletConv2D_69784628625750
// MI455X (gfx1250) — compile-verified
//
#include <hip/hip_runtime.h>

#define BATCH 4
#define CH    64
#define HGT   256
#define WID   256
#define HW    (HGT*WID)
#define NPLANE ((size_t)BATCH*CH*HW)   // 16,777,216 floats

typedef float floatx2 __attribute__((ext_vector_type(2)));
typedef float floatx8 __attribute__((ext_vector_type(8)));

#define DEC_LO0 (-0.12940952255092145f)
#define DEC_LO1 ( 0.22414386804185735f)
#define DEC_LO2 ( 0.836516303737469f)
#define DEC_LO3 ( 0.48296291314469025f)
#define DEC_HI0 (-0.48296291314469025f)
#define DEC_HI1 ( 0.836516303737469f)
#define DEC_HI2 (-0.22414386804185735f)
#define DEC_HI3 (-0.12940952255092145f)

__device__ __forceinline__ float soft_thr(float x, float t) {
    float a = fabsf(x) - t;
    a = a > 0.0f ? a : 0.0f;
    return copysignf(a, x);
}

// ---------------------------------------------------------------------------
// Kernel 1: channel mix  xm[b,o,h,w] = sum_c W[o,c] * x[b,c,h,w]
// WMMA f32 16x16x4 (M=16 consecutive w, N=16 out channels, K=64 in 16 steps).
// A-tile staged into LDS with CDNA5 async global->LDS copies (ASYNCcnt).
// ---------------------------------------------------------------------------
__global__ __launch_bounds__(128)
void mix_wmma(const float* __restrict__ x,
              const float* __restrict__ wmat,   // [64][64] row-major (o,c)
              float* __restrict__ xm)
{
    __shared__ float Atile[CH * 16];            // [c][m], 4 KB

    const int bh   = blockIdx.x;
    const int b    = bh / HGT;
    const int h    = bh % HGT;
    const int t    = threadIdx.x;
    const int wave = t >> 5;
    const int lane = t & 31;
    const int m     = lane & 15;
    const int khalf = lane >> 4;
    const int nb    = wave * 16;

    const size_t base_in = (size_t)b * CH * HW + (size_t)h * WID;

    // Preload B fragments for all 16 K-steps.
    floatx2 Bf[16];
    {
        const float* wrow = wmat + (size_t)(nb + m) * CH;
        #pragma unroll
        for (int ks = 0; ks < 16; ++ks) {
            int k = ks * 4 + khalf * 2;
            Bf[ks].x = wrow[k];
            Bf[ks].y = wrow[k + 1];
        }
    }

    const int c    = t >> 1;
    const int half = t & 1;
    // flat LDS pointer low 32 bits == raw LDS byte offset
    const unsigned lds0 = (unsigned)(size_t)(&Atile[c * 16 + half * 8]);

    for (int tile = 0; tile < WID / 16; ++tile) {
        const int w0 = tile * 16;
        {
            const float* gsrc = x + base_in + (size_t)c * HW + w0 + half * 8;
            unsigned long long ga = (unsigned long long)(size_t)gsrc;
            // async copy 2 x 16B: same inst offset applies to LDS and mem addr
            asm volatile(
                "global_load_async_to_lds_b128 %0, %1, off\n\t"
                "global_load_async_to_lds_b128 %0, %1, off offset:16\n\t"
                "s_wait_asynccnt 0x0"
                :
                : "v"(lds0), "v"(ga)
                : "memory");
        }
        __syncthreads();

        floatx8 acc = {};
        #pragma unroll
        for (int ks = 0; ks < 16; ++ks) {
            int k = ks * 4 + khalf * 2;
            floatx2 a;
            a.x = Atile[(k    ) * 16 + m];
            a.y = Atile[(k + 1) * 16 + m];
            acc = __builtin_amdgcn_wmma_f32_16x16x4_f32(
                false, a, false, Bf[ks], (short)0, acc, false, false);
        }

        float* dst = xm + base_in + (size_t)(nb + m) * HW + w0 + khalf * 8;
        *(float4*)(dst)     = *((float4*)&acc);
        *(float4*)(dst + 4) = *((float4*)&acc + 1);
        __syncthreads();
    }
}

// ---------------------------------------------------------------------------
// Kernel 2: fully fused per-channel SWT pipeline on a 32x32 output tile:
//   S1 (d=1, analysis) -> threshold detail D1
//   S2 (d=2, analysis) -> threshold ll (T2) + detail D2
//   R2 (d=2, synthesis) -> REC
//   R1 (d=1, synthesis) -> out = base + result
// Receptive field of out on xm is [-9,+9] per axis -> 50x50 halo tile in LDS.
// All intermediates stay in LDS (60.4 KB with liveness-based reuse).
// ---------------------------------------------------------------------------
#define TS    32
#define XMD   50   // xm tile:   rows [h0-9 .. h0+40]
#define LLD   47   // LL1:       rows [h0-9 .. h0+37]
#define D1D   35   // D1 / REC:  rows [h0-3 .. h0+31]
#define T2D   41   // T2/D2/tau: rows [h0-9 .. h0+31]

__global__ __launch_bounds__(256)
void fused_swt(const float* __restrict__ xm,
               const float* __restrict__ tau,   // this pod's [H][W]
               const float* __restrict__ v1,    // 3 scales, level 1
               const float* __restrict__ v2,    // 3 scales, level 2
               const float* __restrict__ base,  // x (pod 0) or out (pod 1)
               float* __restrict__ out)
{
    __shared__ float smem[15108];               // 60,432 B
    float* sXM  = smem;                         // 2500  [A-B]
    float* sLL1 = smem + 2500;                  // 2209  [B-C]
    float* sD1  = smem + 4709;                  // 3*1225 [B-E]
    float* sTAU = smem + 8384;                  // 1681  [A-C]
    float* sD2  = smem + 10065;                 // 3*1681 [C-D]
    float* sT2  = sXM;                          // 1681  [C-D] (reuse XM)
    float* sREC = sLL1;                         // 1225  [D-E] (reuse LL1)

    const float LO[4] = {DEC_LO0, DEC_LO1, DEC_LO2, DEC_LO3};
    const float HI[4] = {DEC_HI0, DEC_HI1, DEC_HI2, DEC_HI3};

    const int t  = threadIdx.x;
    const int n  = blockIdx.x >> 6;             // plane 0..255
    const int tl = blockIdx.x & 63;             // 8x8 tiles
    const int h0 = (tl >> 3) * TS;
    const int w0 = (tl & 7) * TS;
    const size_t pbase = (size_t)n * HW;

    const float v10 = v1[0], v11 = v1[1], v12 = v1[2];
    const float v20 = v2[0], v21 = v2[1], v22 = v2[2];

    // ---- Stage A: load xm halo tile + tau tile (wrap addressing) ----------
    for (int e = t; e < XMD * XMD; e += 256) {
        int i = e / XMD, j = e % XMD;
        int gh = (h0 - 9 + i) & (HGT - 1);
        int gw = (w0 - 9 + j) & (WID - 1);
        sXM[e] = xm[pbase + (size_t)gh * WID + gw];
    }
    for (int e = t; e < T2D * T2D; e += 256) {
        int i = e / T2D, j = e % T2D;
        int gh = (h0 - 9 + i) & (HGT - 1);
        int gw = (w0 - 9 + j) & (WID - 1);
        sTAU[e] = tau[(size_t)gh * WID + gw];
    }
    __syncthreads();

    // ---- Stage B: level-1 analysis (d=1): LL1 47x47, D1 thresholded 35x35 -
    for (int e = t; e < LLD * LLD; e += 256) {
        int i = e / LLD, j = e % LLD;
        float sLo[4], sHi[4];
        #pragma unroll
        for (int a = 0; a < 4; ++a) {
            const float* row = sXM + (i + a) * XMD + j;
            float rl = 0.0f, rh = 0.0f;
            #pragma unroll
            for (int bb = 0; bb < 4; ++bb) {
                float xv = row[bb];
                rl = fmaf(LO[bb], xv, rl);
                rh = fmaf(HI[bb], xv, rh);
            }
            sLo[a] = rl; sHi[a] = rh;
        }
        float ll = 0.0f, lh = 0.0f, hl = 0.0f, hh = 0.0f;
        #pragma unroll
        for (int a = 0; a < 4; ++a) {
            ll = fmaf(LO[a], sLo[a], ll);
            lh = fmaf(LO[a], sHi[a], lh);
            hl = fmaf(HI[a], sLo[a], hl);
            hh = fmaf(HI[a], sHi[a], hh);
        }
        sLL1[e] = ll;
        if (i >= 6 && i <= 40 && j >= 6 && j <= 40) {
            float t0 = sTAU[i * T2D + j];       // same (h0-9)-based frame
            int e2 = (i - 6) * D1D + (j - 6);
            sD1[e2]        = soft_thr(v10 * lh, t0);
            sD1[1225 + e2] = soft_thr(v11 * hl, t0);
            sD1[2450 + e2] = soft_thr(v12 * hh, t0);
        }
    }
    __syncthreads();

    // ---- Stage C: level-2 analysis (d=2): T2 + D2, all thresholded, 41x41 -
    for (int e = t; e < T2D * T2D; e += 256) {
        int q = e / T2D, j = e % T2D;
        float sLo[4], sHi[4];
        #pragma unroll
        for (int a = 0; a < 4; ++a) {
            const float* row = sLL1 + (q + 2 * a) * LLD + j;
            float rl = 0.0f, rh = 0.0f;
            #pragma unroll
            for (int bb = 0; bb < 4; ++bb) {
                float xv = row[2 * bb];
                rl = fmaf(LO[bb], xv, rl);
                rh = fmaf(HI[bb], xv, rh);
            }
            sLo[a] = rl; sHi[a] = rh;
        }
        float ll = 0.0f, lh = 0.0f, hl = 0.0f, hh = 0.0f;
        #pragma unroll
        for (int a = 0; a < 4; ++a) {
            ll = fmaf(LO[a], sLo[a], ll);
            lh = fmaf(LO[a], sHi[a], lh);
            hl = fmaf(HI[a], sLo[a], hl);
            hh = fmaf(HI[a], sHi[a], hh);
        }
        float t0 = sTAU[e];
        sT2[e]        = soft_thr(ll, t0);
        sD2[e]        = soft_thr(v20 * lh, t0);
        sD2[1681 + e] = soft_thr(v21 * hl, t0);
        sD2[3362 + e] = soft_thr(v22 * hh, t0);
    }
    __syncthreads();

    // ---- Stage D: level-2 synthesis (d=2): REC 35x35 -----------------------
    for (int e = t; e < D1D * D1D; e += 256) {
        int p = e / D1D, j = e % D1D;
        float acc = 0.0f;
        #pragma unroll
        for (int a = 0; a < 4; ++a) {
            int ia = p + 6 - 2 * a;             // frame base h0-9
            float rl = 0.0f, rh = 0.0f;
            #pragma unroll
            for (int bb = 0; bb < 4; ++bb) {
                int idx = ia * T2D + (j + 6 - 2 * bb);
                rl = fmaf(LO[bb], sT2[idx], rl);
                rl = fmaf(HI[bb], sD2[idx], rl);
                rh = fmaf(LO[bb], sD2[1681 + idx], rh);
                rh = fmaf(HI[bb], sD2[3362 + idx], rh);
            }
            acc = fmaf(LO[a], rl, acc);
            acc = fmaf(HI[a], rh, acc);
        }
        sREC[e] = 0.25f * acc;
    }
    __syncthreads();

    // ---- Stage E: level-1 synthesis (d=1): out 32x32 + base ---------------
    for (int e = t; e < TS * TS; e += 256) {
        int o = e / TS, j = e % TS;
        float acc = 0.0f;
        #pragma unroll
        for (int a = 0; a < 4; ++a) {
            int ia = o + 3 - a;                 // frame base h0-3
            float rl = 0.0f, rh = 0.0f;
            #pragma unroll
            for (int bb = 0; bb < 4; ++bb) {
                int idx = ia * D1D + (j + 3 - bb);
                rl = fmaf(LO[bb], sREC[idx], rl);
                rl = fmaf(HI[bb], sD1[idx], rl);
                rh = fmaf(LO[bb], sD1[1225 + idx], rh);
                rh = fmaf(HI[bb], sD1[2450 + idx], rh);
            }
            acc = fmaf(LO[a], rl, acc);
            acc = fmaf(HI[a], rh, acc);
        }
        size_t g = pbase + (size_t)(h0 + o) * WID + (w0 + j);
        out[g] = base[g] + 0.25f * acc;
    }
}

// ---------------------------------------------------------------------------
// Per pod i (re-associated: mix commutes with the per-channel spatial SWT):
//   1. XM = W_i @ x                       (WMMA)
//   2. fused S1->S2->R2->R1, out = base + recon
// ---------------------------------------------------------------------------
extern "C" void kernel_launch(void* const* d_in, const int* in_sizes, int n_in,
                              void* d_out, int out_size, void* d_ws, size_t ws_size,
                              hipStream_t stream)
{
    const float* x      = (const float*)d_in[0];
    const float* conv_w = (const float*)d_in[1];  // [2][64][64]
    const float* v      = (const float*)d_in[2];  // [2][2][3]
    const float* tau    = (const float*)d_in[3];  // [2][256][256]
    float* out = (float*)d_out;
    float* XM  = (float*)d_ws;                    // one plane set: 67 MB

    for (int i = 0; i < 2; ++i) {
        mix_wmma<<<BATCH * HGT, 128, 0, stream>>>(
            x, conv_w + (size_t)i * CH * CH, XM);

        fused_swt<<<BATCH * CH * 64, 256, 0, stream>>>(
            XM, tau + (size_t)i * HW,
            v + (i * 2 + 0) * 3, v + (i * 2 + 1) * 3,
            (i == 0 ? x : (const float*)out), out);
    }
}